// PureTriadicBrain_73753178407524
// MI455X (gfx1250) — compile-verified
//
#include <hip/hip_runtime.h>
#include <cfloat>
#include <cmath>

typedef __attribute__((ext_vector_type(2))) float v2f;
typedef __attribute__((ext_vector_type(8))) float v8f;

constexpr int   kB = 4;
constexpr int   kN = 512;
constexpr int   kD = 128;
constexpr float kBound = 1.8477590650225735f;   // sqrt(2 + sqrt(2))
constexpr float kInvSqrtD = 0.08838834764831845f; // 1/sqrt(128)

// ---------------------------------------------------------------------------
// small helpers
// ---------------------------------------------------------------------------
__global__ void k_zero(float* __restrict__ p, int n) {
  int i = blockIdx.x * blockDim.x + threadIdx.x;
  if (i < n) p[i] = 0.0f;
}

// ---------------------------------------------------------------------------
// 1) per-node matvec + buntanh:  y[b,n,e] = buntanh(sum_d state[b,n,d]*w[b,n,d,e])
//    one 128-thread block per (b,n); state row staged in LDS; coalesced w reads
// ---------------------------------------------------------------------------
__global__ void k_matvec_buntanh(const float* __restrict__ state,
                                 const float* __restrict__ w,
                                 const float* __restrict__ hp_slope,
                                 float* __restrict__ out) {
  __shared__ float s[kD];
  int bn = blockIdx.x;
  int b  = bn / kN;
  int e  = threadIdx.x;
  const float* srow = state + (size_t)bn * kD;
  s[e] = srow[e];
  __syncthreads();
  const float* wm = w + (size_t)bn * kD * kD;
  float acc = 0.0f;
#pragma unroll 8
  for (int d = 0; d < kD; ++d)
    acc = fmaf(s[d], wm[(size_t)d * kD + e], acc);
  float slope = hp_slope[b];
  out[(size_t)bn * kD + e] = tanhf(acc - tanhf(acc) * slope) * kBound;
}

// ---------------------------------------------------------------------------
// 2) raw_A[b,n,m] = dot(Q[b,n,:], K[b,m,:]) / sqrt(D)  via V_WMMA_F32_16X16X4_F32
//    one wave32 per 16x16 output tile.
//    f32 A-frag layout (ISA 7.12.2): lanes 0-15 -> M=lane, K=k0..k0+1 in v0,v1;
//    lanes 16-31 -> M=lane-16, K=k0+2..k0+3. B mirrors with N=lane&15.
// ---------------------------------------------------------------------------
__global__ void k_gemm_qk(const float* __restrict__ Q,
                          const float* __restrict__ K,
                          float* __restrict__ rawA) {
  int mt = blockIdx.x, nt = blockIdx.y, b = blockIdx.z;
  int lane = threadIdx.x;
  int r16 = lane & 15;
  int kh  = (lane >> 4) * 2;
  const float* Qb = Q + (size_t)b * kN * kD;
  const float* Kb = K + (size_t)b * kN * kD;
  v8f c = {};
  for (int k0 = 0; k0 < kD; k0 += 4) {
    v2f a, bb;
    a.x  = Qb[(size_t)(nt * 16 + r16) * kD + k0 + kh];
    a.y  = Qb[(size_t)(nt * 16 + r16) * kD + k0 + kh + 1];
    bb.x = Kb[(size_t)(mt * 16 + r16) * kD + k0 + kh];
    bb.y = Kb[(size_t)(mt * 16 + r16) * kD + k0 + kh + 1];
    c = __builtin_amdgcn_wmma_f32_16x16x4_f32(false, a, false, bb,
                                              (short)0, c, false, false);
  }
  float* Ab = rawA + (size_t)b * kN * kN;
  int rb = (lane >> 4) * 8;
#pragma unroll
  for (int r = 0; r < 8; ++r)
    Ab[(size_t)(nt * 16 + rb + r) * kN + mt * 16 + r16] = c[r] * kInvSqrtD;
}

// ---------------------------------------------------------------------------
// 3) per-batch sum of squares (for rms_norm over (-1,-2))
// ---------------------------------------------------------------------------
__global__ void k_sumsq_perb(const float* __restrict__ x, int per_b,
                             float* __restrict__ out) {
  __shared__ double sred[256];
  int b = blockIdx.x;
  const float* xb = x + (size_t)b * per_b;
  double s = 0.0;
  for (int i = threadIdx.x; i < per_b; i += blockDim.x) {
    double v = xb[i];
    s += v * v;
  }
  sred[threadIdx.x] = s;
  __syncthreads();
  for (int st = 128; st > 0; st >>= 1) {
    if (threadIdx.x < st) sred[threadIdx.x] += sred[threadIdx.x + st];
    __syncthreads();
  }
  if (threadIdx.x == 0) out[b] = (float)sred[0];
}

// ---------------------------------------------------------------------------
// 4) A_ema_new = (rawA_rms * isf + (A_ema*cm + rawA_rms*(1-cm))*(1-isf)) * mask
// ---------------------------------------------------------------------------
__global__ void k_aema(const float* __restrict__ rawA,
                       const float* __restrict__ ssq,
                       const float* __restrict__ A_ema,
                       const float* __restrict__ hp_cm,
                       const int* __restrict__ step,
                       float* __restrict__ outA) {
  size_t idx = (size_t)blockIdx.x * blockDim.x + threadIdx.x;
  size_t tot = (size_t)kB * kN * kN;
  if (idx >= tot) return;
  int b = (int)(idx / ((size_t)kN * kN));
  int n = (int)((idx / kN) % kN);
  float inv = rsqrtf(ssq[b] / (float)(kN * kN) + 1e-8f);
  float rn = rawA[idx] * inv;
  float t = (float)step[0] + 1.0f;
  float isf = (t == 1.0f) ? 1.0f : 0.0f;
  float cm = hp_cm[b];
  float v = rn * isf + (A_ema[idx] * cm + rn * (1.0f - cm)) * (1.0f - isf);
  outA[idx] = (n >= 2) ? v : 0.0f;
}

// ---------------------------------------------------------------------------
// 5) sinkhorn: row/col logsumexp + exp
// ---------------------------------------------------------------------------
__global__ void k_row_lse(const float* __restrict__ la,
                          const float* __restrict__ v,
                          float* __restrict__ u) {
  __shared__ float sred[128];
  int bn = blockIdx.x;
  int b  = bn / kN;
  const float* row = la + (size_t)bn * kN;
  const float* vb  = v + (size_t)b * kN;
  int tid = threadIdx.x;
  float vals[4];
  float mx = -FLT_MAX;
#pragma unroll
  for (int c = 0; c < 4; ++c) {
    int m = tid + c * 128;
    vals[c] = row[m] + vb[m];
    mx = fmaxf(mx, vals[c]);
  }
  sred[tid] = mx; __syncthreads();
  for (int st = 64; st > 0; st >>= 1) {
    if (tid < st) sred[tid] = fmaxf(sred[tid], sred[tid + st]);
    __syncthreads();
  }
  mx = sred[0]; __syncthreads();
  float s = 0.0f;
#pragma unroll
  for (int c = 0; c < 4; ++c) s += expf(vals[c] - mx);
  sred[tid] = s; __syncthreads();
  for (int st = 64; st > 0; st >>= 1) {
    if (tid < st) sred[tid] += sred[tid + st];
    __syncthreads();
  }
  if (tid == 0) u[bn] = -(mx + logf(sred[0]));
}

__global__ void k_col_lse(const float* __restrict__ la,
                          const float* __restrict__ u,
                          float* __restrict__ v) {
  __shared__ float sred[128];
  int bm = blockIdx.x;
  int b  = bm / kN;
  int m  = bm % kN;
  const float* lab = la + (size_t)b * kN * kN;
  const float* ub  = u + (size_t)b * kN;
  int tid = threadIdx.x;
  float vals[4];
  float mx = -FLT_MAX;
#pragma unroll
  for (int c = 0; c < 4; ++c) {
    int n = tid + c * 128;
    vals[c] = lab[(size_t)n * kN + m] + ub[n];
    mx = fmaxf(mx, vals[c]);
  }
  sred[tid] = mx; __syncthreads();
  for (int st = 64; st > 0; st >>= 1) {
    if (tid < st) sred[tid] = fmaxf(sred[tid], sred[tid + st]);
    __syncthreads();
  }
  mx = sred[0]; __syncthreads();
  float s = 0.0f;
#pragma unroll
  for (int c = 0; c < 4; ++c) s += expf(vals[c] - mx);
  sred[tid] = s; __syncthreads();
  for (int st = 64; st > 0; st >>= 1) {
    if (tid < st) sred[tid] += sred[tid + st];
    __syncthreads();
  }
  if (tid == 0) v[bm] = -(mx + logf(sred[0]));
}

__global__ void k_sink_exp(const float* __restrict__ la,
                           const float* __restrict__ u,
                           const float* __restrict__ v,
                           float* __restrict__ out, int apply_mask) {
  size_t idx = (size_t)blockIdx.x * blockDim.x + threadIdx.x;
  size_t tot = (size_t)kB * kN * kN;
  if (idx >= tot) return;
  int b = (int)(idx / ((size_t)kN * kN));
  int n = (int)((idx / kN) % kN);
  int m = (int)(idx % kN);
  float r = expf(la[idx] + u[(size_t)b * kN + n] + v[(size_t)b * kN + m]);
  if (apply_mask && n < 2) r = 0.0f;
  out[idx] = r;
}

__global__ void k_sparse(const float* __restrict__ rA, float* __restrict__ sA) {
  size_t idx = (size_t)blockIdx.x * blockDim.x + threadIdx.x;
  size_t tot = (size_t)kB * kN * kN;
  if (idx >= tot) return;
  float v = rA[idx];
  sA[idx] = (v > (0.5f / (float)kN)) ? v : 0.0f;
}

// ---------------------------------------------------------------------------
// 6) unbiased variance of A per batch -> vd = relu(hp_a_variance - var)
// ---------------------------------------------------------------------------
__global__ void k_var_vd(const float* __restrict__ A,
                         const float* __restrict__ hp_av,
                         float* __restrict__ vd) {
  __shared__ double sred[512];
  int b = blockIdx.x;
  const int M = kN * kN;
  const float* Ab = A + (size_t)b * M;
  double s = 0.0, s2 = 0.0;
  for (int i = threadIdx.x; i < M; i += blockDim.x) {
    double v = Ab[i];
    s += v; s2 += v * v;
  }
  sred[threadIdx.x] = s;
  sred[256 + threadIdx.x] = s2;
  __syncthreads();
  for (int st = 128; st > 0; st >>= 1) {
    if (threadIdx.x < st) {
      sred[threadIdx.x] += sred[threadIdx.x + st];
      sred[256 + threadIdx.x] += sred[256 + threadIdx.x + st];
    }
    __syncthreads();
  }
  if (threadIdx.x == 0) {
    double var = (sred[256] - sred[0] * sred[0] / M) / (double)(M - 1);
    float d = hp_av[b] - (float)var;
    vd[b] = d > 0.0f ? d : 0.0f;
  }
}

// ---------------------------------------------------------------------------
// 7) target = softsign(A @ (output + vd*bnoise))  via WMMA f32 16x16x4
// ---------------------------------------------------------------------------
__global__ void k_gemm_target(const float* __restrict__ A,
                              const float* __restrict__ outp,
                              const float* __restrict__ bnoise,
                              const float* __restrict__ vd,
                              float* __restrict__ target) {
  int dt = blockIdx.x, nt = blockIdx.y, b = blockIdx.z;
  int lane = threadIdx.x;
  int r16 = lane & 15;
  int kh  = (lane >> 4) * 2;
  const float* Ab = A + (size_t)b * kN * kN;
  const float* Ob = outp + (size_t)b * kN * kD;
  const float* Bn = bnoise + (size_t)b * kN * kD;
  float vdb = vd[b];
  v8f c = {};
  for (int k0 = 0; k0 < kN; k0 += 4) {
    v2f a, bb;
    a.x = Ab[(size_t)(nt * 16 + r16) * kN + k0 + kh];
    a.y = Ab[(size_t)(nt * 16 + r16) * kN + k0 + kh + 1];
    size_t i0 = (size_t)(k0 + kh) * kD + dt * 16 + r16;
    size_t i1 = (size_t)(k0 + kh + 1) * kD + dt * 16 + r16;
    bb.x = Ob[i0] + vdb * Bn[i0];
    bb.y = Ob[i1] + vdb * Bn[i1];
    c = __builtin_amdgcn_wmma_f32_16x16x4_f32(false, a, false, bb,
                                              (short)0, c, false, false);
  }
  float* Tb = target + (size_t)b * kN * kD;
  int rb = (lane >> 4) * 8;
#pragma unroll
  for (int r = 0; r < 8; ++r) {
    float v = c[r];
    v = v / (1.0f + fabsf(v));
    Tb[(size_t)(nt * 16 + rb + r) * kD + dt * 16 + r16] = v;
  }
}

__global__ void k_set_rows(const float* __restrict__ eye,
                           const float* __restrict__ stomach,
                           float* __restrict__ target) {
  int b = blockIdx.x, d = threadIdx.x;
  target[((size_t)b * kN + 0) * kD + d] = eye[(size_t)b * kD + d];
  target[((size_t)b * kN + 1) * kD + d] = stomach[(size_t)b * kD + d];
}

// ---------------------------------------------------------------------------
// 8) error chain (elementwise)
// ---------------------------------------------------------------------------
__global__ void k_errs(const float* __restrict__ pred,
                       const float* __restrict__ target,
                       const float* __restrict__ q,
                       const float* __restrict__ k,
                       float* __restrict__ e1, float* __restrict__ e2,
                       float* __restrict__ e3) {
  size_t idx = (size_t)blockIdx.x * blockDim.x + threadIdx.x;
  size_t tot = (size_t)kB * kN * kD;
  if (idx >= tot) return;
  float p = pred[idx], tg = target[idx];
  float e1b = p - tg;
  float e2v = k[idx] - e1b;
  float e3v = q[idx] - e2v;
  e1[idx] = p - (tg + e3v);
  e2[idx] = e2v;
  e3[idx] = e3v;
}

// ---------------------------------------------------------------------------
// 9) per-batch sum |state| ; global unbiased std of err
// ---------------------------------------------------------------------------
__global__ void k_state_mag(const float* __restrict__ state,
                            float* __restrict__ mag) {
  __shared__ double sred[256];
  int b = blockIdx.x;
  const int M = kN * kD;
  const float* sb = state + (size_t)b * M;
  double s = 0.0;
  for (int i = threadIdx.x; i < M; i += blockDim.x) s += fabsf(sb[i]);
  sred[threadIdx.x] = s; __syncthreads();
  for (int st = 128; st > 0; st >>= 1) {
    if (threadIdx.x < st) sred[threadIdx.x] += sred[threadIdx.x + st];
    __syncthreads();
  }
  if (threadIdx.x == 0) mag[b] = (float)sred[0];
}

__global__ void k_err_std(const float* __restrict__ e, float* __restrict__ out) {
  __shared__ double sred[512];
  const int M = kB * kN * kD;
  double s = 0.0, s2 = 0.0;
  for (int i = threadIdx.x; i < M; i += blockDim.x) {
    double v = e[i];
    s += v; s2 += v * v;
  }
  sred[threadIdx.x] = s;
  sred[256 + threadIdx.x] = s2;
  __syncthreads();
  for (int st = 128; st > 0; st >>= 1) {
    if (threadIdx.x < st) {
      sred[threadIdx.x] += sred[threadIdx.x + st];
      sred[256 + threadIdx.x] += sred[256 + threadIdx.x + st];
    }
    __syncthreads();
  }
  if (threadIdx.x == 0) {
    double var = (sred[256] - sred[0] * sred[0] / M) / (double)(M - 1);
    out[0] = (float)sqrt(var > 0.0 ? var : 0.0);
  }
}

// ---------------------------------------------------------------------------
// 10) fused piece_grad + piece_step:
//     one 256-thread block per (b,n); streams the 128x128 tile once;
//     w_tmp cached in 64 regs/thread; Σw² block-reduced, then rescaled store.
// ---------------------------------------------------------------------------
__global__ void __launch_bounds__(256)
k_piece_step(const float* __restrict__ err,  const float* __restrict__ Ebase,
             const float* __restrict__ n1,   const float* __restrict__ n2,
             const float* __restrict__ state,const float* __restrict__ w,
             const float* __restrict__ g,    const float* __restrict__ mag,
             const float* __restrict__ estd,
             const float* __restrict__ hp_es, const float* __restrict__ hp_ns,
             const float* __restrict__ hp_rs, const float* __restrict__ hp_wd,
             const float* __restrict__ hp_sp, const float* __restrict__ hp_wi,
             const float* __restrict__ hp_mo, const float* __restrict__ hp_lr,
             const float* __restrict__ hp_ws, const int* __restrict__ step,
             float* __restrict__ outE, float* __restrict__ outW,
             float* __restrict__ outG) {
  __shared__ float s_err[kD], s_plast[kD], s_noisy[kD], s_kern[kD];
  __shared__ float sred[256];
  __shared__ float s_mnsq;
  int bn = blockIdx.x;
  int b  = bn / kN;
  int tid = threadIdx.x;
  size_t roff = (size_t)bn * kD;
  float es = hp_es[b], nsv = hp_ns[b], rsv = hp_rs[b], wdv = hp_wd[b];
  float spv = hp_sp[b], wiv = hp_wi[b], mov = hp_mo[b], lrv = hp_lr[b];
  float wsv = hp_ws[b];
  float t = (float)step[0] + 1.0f;

  float ev = (tid < kD) ? err[roff + tid] : 0.0f;
  // row softmax: max
  sred[tid] = (tid < kD) ? ev : -FLT_MAX;
  __syncthreads();
  for (int st = 128; st > 0; st >>= 1) {
    if (tid < st) sred[tid] = fmaxf(sred[tid], sred[tid + st]);
    __syncthreads();
  }
  float mx = sred[0]; __syncthreads();
  float ex = (tid < kD) ? expf(ev - mx) : 0.0f;
  sred[tid] = ex; __syncthreads();
  for (int st = 128; st > 0; st >>= 1) {
    if (tid < st) sred[tid] += sred[tid + st];
    __syncthreads();
  }
  float sm = sred[0]; __syncthreads();

  if (tid < kD) {
    float ecur = ex / sm;
    float enew = Ebase[roff + tid] * (1.0f - es) + es * ecur;
    outE[roff + tid] = enew;
    float z = (ecur - enew) * rsv;                  // advantage * rew_sens
    s_plast[tid] = 1.0f + z * rsqrtf(z * z + 1e-8f);
    s_err[tid] = ev;
    float noisy = state[roff + tid] + n1[roff + tid] / (1.0f + mag[b]) +
                  n2[roff + tid] * estd[0] * nsv;
    s_noisy[tid] = noisy;
    float co = fmodf(t * spv, (float)kD);
    float center = co * (float)kD;
    float sig = wiv * (float)kD;
    float dif = fabsf((float)tid - center);
    dif = fminf(dif, (float)kD - dif);
    s_kern[tid] = expf(-(dif * dif) / (2.0f * sig * sig));
  }
  __syncthreads();
  // mean_j noisy^2
  sred[tid] = (tid < kD) ? s_noisy[tid] * s_noisy[tid] : 0.0f;
  __syncthreads();
  for (int st = 128; st > 0; st >>= 1) {
    if (tid < st) sred[tid] += sred[tid + st];
    __syncthreads();
  }
  if (tid == 0) s_mnsq = sred[0] / (float)kD;
  __syncthreads();
  float mnsq = s_mnsq;

  size_t mbase = (size_t)bn * kD * kD;
  float wreg[64];
  float ssq = 0.0f;
#pragma unroll
  for (int kk = 0; kk < 64; ++kk) {
    int e = tid + kk * 256;
    int i = e >> 7;
    int j = e & (kD - 1);
    float wv = w[mbase + e];
    float gv = g[mbase + e];
    float ei = s_err[i];
    float lg = -(ei * s_noisy[j]) * rsqrtf(ei * ei * mnsq + 1e-8f);
    float grad = s_plast[i] * lg - wdv * wv;
    grad *= s_kern[j];
    float gn = mov * gv + (1.0f - mov) * grad;
    outG[mbase + e] = gn;
    float wt = wv + lrv * gn;
    wreg[kk] = wt;
    ssq += wt * wt;
  }
  sred[tid] = ssq; __syncthreads();
  for (int st = 128; st > 0; st >>= 1) {
    if (tid < st) sred[tid] += sred[tid + st];
    __syncthreads();
  }
  float scale = rsqrtf(sred[0] / (float)(kD * kD) + 1e-8f) * wsv;
#pragma unroll
  for (int kk = 0; kk < 64; ++kk)
    outW[mbase + tid + (size_t)kk * 256] = wreg[kk] * scale;
}

// ---------------------------------------------------------------------------
// host launcher
// ---------------------------------------------------------------------------
extern "C" void kernel_launch(void* const* d_in, const int* in_sizes, int n_in,
                              void* d_out, int out_size, void* d_ws,
                              size_t ws_size, hipStream_t stream) {
  (void)in_sizes; (void)n_in; (void)out_size; (void)ws_size;
  const float* state   = (const float*)d_in[0];
  const float* output  = (const float*)d_in[1];
  const float* A_ema   = (const float*)d_in[2];
  const float* eye     = (const float*)d_in[3];
  const float* stomach = (const float*)d_in[4];
  const float* w1 = (const float*)d_in[5];
  const float* w2 = (const float*)d_in[6];
  const float* w3 = (const float*)d_in[7];
  const float* g1 = (const float*)d_in[8];
  const float* g2 = (const float*)d_in[9];
  const float* g3 = (const float*)d_in[10];
  const float* E1 = (const float*)d_in[11];
  const float* E2 = (const float*)d_in[12];
  const float* E3 = (const float*)d_in[13];
  const float* bnoise = (const float*)d_in[14];
  const float* n1a = (const float*)d_in[15];
  const float* n2a = (const float*)d_in[16];
  const float* n1b = (const float*)d_in[17];
  const float* n2b = (const float*)d_in[18];
  const float* n1c = (const float*)d_in[19];
  const float* n2c = (const float*)d_in[20];
  const float* hp_un = (const float*)d_in[21];
  const float* hp_es = (const float*)d_in[22];
  const float* hp_ns = (const float*)d_in[23];
  const float* hp_rs = (const float*)d_in[24];
  const float* hp_wd = (const float*)d_in[25];
  const float* hp_sp = (const float*)d_in[26];
  const float* hp_wi = (const float*)d_in[27];
  const float* hp_mo = (const float*)d_in[28];
  const float* hp_lr = (const float*)d_in[29];
  const float* hp_ws = (const float*)d_in[30];
  const float* hp_cm = (const float*)d_in[31];
  const float* hp_av = (const float*)d_in[32];
  const int*   stepc = (const int*)d_in[33];

  const size_t BND  = (size_t)kB * kN * kD;
  const size_t BNN  = (size_t)kB * kN * kN;
  const size_t BNDD = (size_t)kB * kN * kD * kD;

  float* out = (float*)d_out;
  float* o_pred   = out;
  float* o_target = out + BND;
  float* o_A      = out + 2 * BND;
  float* o_Aema   = o_A + BNN;
  float* o_w1 = o_Aema + BNN;
  float* o_w2 = o_w1 + BNDD;
  float* o_w3 = o_w2 + BNDD;
  float* o_g1 = o_w3 + BNDD;
  float* o_g2 = o_g1 + BNDD;
  float* o_g3 = o_g2 + BNDD;
  float* o_E1 = o_g3 + BNDD;
  float* o_E2 = o_E1 + BND;
  float* o_E3 = o_E2 + BND;

  float* ws   = (float*)d_ws;
  float* q    = ws;
  float* kbuf = q + BND;
  float* rawA = kbuf + BND;
  float* rA   = rawA + BNN;
  float* su   = rA + BNN;
  float* sv   = su + (size_t)kB * kN;
  float* e1   = sv + (size_t)kB * kN;
  float* e2   = e1 + BND;
  float* e3   = e2 + BND;
  float* sc   = e3 + BND;     // scalar slots
  float* mag4 = sc;           // [4]
  float* ssqA = sc + 4;       // [4]
  float* vd4  = sc + 8;       // [4]
  float* std1 = sc + 12;
  float* std2 = sc + 13;
  float* std3 = sc + 14;

  // 1. matvecs + buntanh (pred<-w1, keys<-w2, queries<-w3)
  k_matvec_buntanh<<<kB * kN, kD, 0, stream>>>(state, w1, hp_un, o_pred);
  k_matvec_buntanh<<<kB * kN, kD, 0, stream>>>(state, w2, hp_un, kbuf);
  k_matvec_buntanh<<<kB * kN, kD, 0, stream>>>(state, w3, hp_un, q);
  // 2. raw_A via WMMA
  k_gemm_qk<<<dim3(kN / 16, kN / 16, kB), 32, 0, stream>>>(q, kbuf, rawA);
  // 3. rms over (-1,-2)
  k_sumsq_perb<<<kB, 256, 0, stream>>>(rawA, kN * kN, ssqA);
  // 4. A_ema_new
  {
    int tot = kB * kN * kN;
    k_aema<<<(tot + 255) / 256, 256, 0, stream>>>(rawA, ssqA, A_ema, hp_cm,
                                                  stepc, o_Aema);
  }
  // 5/7. sinkhorn twice
  auto run_sink = [&](const float* la, float* res, int mask) {
    int un = 2 * kB * kN;
    k_zero<<<(un + 255) / 256, 256, 0, stream>>>(su, un);  // zero u and v
    for (int it = 0; it < 5; ++it) {
      k_row_lse<<<kB * kN, 128, 0, stream>>>(la, sv, su);
      k_col_lse<<<kB * kN, 128, 0, stream>>>(la, su, sv);
    }
    int tot = kB * kN * kN;
    k_sink_exp<<<(tot + 255) / 256, 256, 0, stream>>>(la, su, sv, res, mask);
  };
  run_sink(o_Aema, rA, 0);
  {
    int tot = kB * kN * kN;
    k_sparse<<<(tot + 255) / 256, 256, 0, stream>>>(rA, rawA);
  }
  run_sink(rawA, o_A, 1);
  // 8. variance -> vd
  k_var_vd<<<kB, 256, 0, stream>>>(o_A, hp_av, vd4);
  // 9. target GEMM via WMMA (+softsign)
  k_gemm_target<<<dim3(kD / 16, kN / 16, kB), 32, 0, stream>>>(o_A, output,
                                                               bnoise, vd4,
                                                               o_target);
  // 10. rows 0/1 <- env
  k_set_rows<<<kB, kD, 0, stream>>>(eye, stomach, o_target);
  // 11. errors
  {
    int tot = kB * kN * kD;
    k_errs<<<(tot + 255) / 256, 256, 0, stream>>>(o_pred, o_target, q, kbuf,
                                                  e1, e2, e3);
  }
  // 12. |state| per batch
  k_state_mag<<<kB, 256, 0, stream>>>(state, mag4);
  // 13. three plastic pieces
  k_err_std<<<1, 256, 0, stream>>>(e1, std1);
  k_piece_step<<<kB * kN, 256, 0, stream>>>(
      e1, E1, n1a, n2a, state, w1, g1, mag4, std1, hp_es, hp_ns, hp_rs, hp_wd,
      hp_sp, hp_wi, hp_mo, hp_lr, hp_ws, stepc, o_E1, o_w1, o_g1);
  k_err_std<<<1, 256, 0, stream>>>(e2, std2);
  k_piece_step<<<kB * kN, 256, 0, stream>>>(
      e2, E2, n1b, n2b, state, w2, g2, mag4, std2, hp_es, hp_ns, hp_rs, hp_wd,
      hp_sp, hp_wi, hp_mo, hp_lr, hp_ws, stepc, o_E2, o_w2, o_g2);
  k_err_std<<<1, 256, 0, stream>>>(e3, std3);
  k_piece_step<<<kB * kN, 256, 0, stream>>>(
      e3, E3, n1c, n2c, state, w3, g3, mag4, std3, hp_es, hp_ns, hp_rs, hp_wd,
      hp_sp, hp_wi, hp_mo, hp_lr, hp_ws, stepc, o_E3, o_w3, o_g3);
}